// ProposalGenerator_62139586839161
// MI455X (gfx1250) — compile-verified
//
#include <hip/hip_runtime.h>
#include <stdint.h>
#include <math.h>

#define NBINS     4096
#define PRE_NMS   6000
#define POST_NMS  1000
#define NMS_T     0.7f
#define MIN_SZ    16.0f
#define IMG_W1    1332.0f   // W-1
#define IMG_H1    799.0f    // H-1
#define PAD       6016      // PRE_NMS padded to 16-row / 32-col tiles
#define WORDS     (PAD/32)  // 188
#define ROWTILES  (PAD/16)  // 376
#define CGRPS     ((WORDS + 7) / 8)   // 24 col-word groups of 8 per block
#define CAP       8192      // candidate sort capacity (64KB LDS)

typedef float v2f __attribute__((ext_vector_type(2)));
typedef float v8f __attribute__((ext_vector_type(8)));
typedef int   v4i __attribute__((vector_size(4 * sizeof(int))));
typedef __attribute__((address_space(1))) v4i* gv4p;   // global
typedef __attribute__((address_space(3))) v4i* lv4p;   // LDS

#if defined(__AMDGCN__) && __has_builtin(__builtin_amdgcn_global_load_async_to_lds_b128)
#define USE_ASYNC_LDS 1
#else
#define USE_ASYNC_LDS 0
#endif

__device__ __forceinline__ void async_copy_b128(const float4* g, float4* l) {
#if USE_ASYNC_LDS
    v4i* gg = (v4i*)g;   // C-style cast drops const, reinterprets
    v4i* ll = (v4i*)l;
    __builtin_amdgcn_global_load_async_to_lds_b128((gv4p)gg, (lv4p)ll, 0, 0);
#else
    *l = *g;
#endif
}

__device__ __forceinline__ void async_copy_wait() {
#if USE_ASYNC_LDS
#if __has_builtin(__builtin_amdgcn_s_wait_asynccnt)
    __builtin_amdgcn_s_wait_asynccnt(0);
#else
    asm volatile("s_wait_asynccnt 0" ::: "memory");
#endif
#endif
}

// ---------------- 0) init: zero histogram + meta ----------------
__global__ void pg_init(unsigned* hist, unsigned* meta) {
    int i = blockIdx.x * blockDim.x + threadIdx.x;
    if (i < NBINS) hist[i] = 0u;
    if (i < 2) meta[i] = 0u;
}

// ---------------- 1) decode anchors + scores + histogram ----------------
__global__ void pg_decode(const float* __restrict__ anchors,
                          const float* __restrict__ deltas,
                          const float* __restrict__ logits,
                          float4* __restrict__ proposals,
                          float* __restrict__ scores,
                          unsigned* __restrict__ hist, int N) {
    __shared__ unsigned sh[NBINS];
    for (int b = threadIdx.x; b < NBINS; b += blockDim.x) sh[b] = 0u;
    __syncthreads();
    int i = blockIdx.x * blockDim.x + threadIdx.x;
    if (i < N) {
        float a0 = anchors[4*i+0], a1 = anchors[4*i+1];
        float a2 = anchors[4*i+2], a3 = anchors[4*i+3];
        float wa = a2 - a0, ha = a3 - a1;
        float xa = a0 + 0.5f*wa, ya = a1 + 0.5f*ha;
        float d0 = deltas[4*i+0], d1 = deltas[4*i+1];
        float d2 = deltas[4*i+2], d3 = deltas[4*i+3];
        float x = d0*wa + xa, y = d1*ha + ya;
        float w = expf(d2)*wa, h = expf(d3)*ha;
        float x1 = fminf(fmaxf(x - 0.5f*w, 0.0f), IMG_W1);
        float y1 = fminf(fmaxf(y - 0.5f*h, 0.0f), IMG_H1);
        float x2 = fminf(fmaxf(x + 0.5f*w, 0.0f), IMG_W1);
        float y2 = fminf(fmaxf(y + 0.5f*h, 0.0f), IMG_H1);
        proposals[i] = make_float4(x1, y1, x2, y2);
        bool valid = (x2 - x1 >= MIN_SZ) && (y2 - y1 >= MIN_SZ);
        float s;
        if (valid) {
            s = 1.0f / (1.0f + expf(-logits[i]));
            int b = (int)(s * (float)NBINS);
            b = b < 0 ? 0 : (b > NBINS-1 ? NBINS-1 : b);
            atomicAdd(&sh[b], 1u);
        } else {
            s = -INFINITY;
        }
        scores[i] = s;
    }
    __syncthreads();
    for (int b = threadIdx.x; b < NBINS; b += blockDim.x)
        if (sh[b]) atomicAdd(&hist[b], sh[b]);
}

// ---------------- 2) threshold bin (suffix scan) ----------------
__global__ void pg_thresh(const unsigned* __restrict__ hist, unsigned* meta) {
    if (threadIdx.x == 0 && blockIdx.x == 0) {
        unsigned acc = 0; int B = 0;
        for (int b = NBINS - 1; b >= 0; --b) {
            acc += hist[b];
            if (acc >= (unsigned)PRE_NMS) { B = b; break; }
        }
        meta[1] = (unsigned)B;
        meta[0] = 0u; // candidate counter
    }
}

// ---------------- 3) compact candidates >= threshold bin ----------------
__global__ void pg_compact(const float* __restrict__ scores,
                           unsigned* __restrict__ meta,
                           unsigned long long* __restrict__ cand, int N) {
    int i = blockIdx.x * blockDim.x + threadIdx.x;
    if (i >= N) return;
    float s = scores[i];
    if (!(s > 0.0f)) return; // invalid -> -inf
    int b = (int)(s * (float)NBINS);
    b = b < 0 ? 0 : (b > NBINS-1 ? NBINS-1 : b);
    if ((unsigned)b >= meta[1]) {
        unsigned p = atomicAdd(&meta[0], 1u);
        if (p < CAP)
            cand[p] = ((unsigned long long)__float_as_uint(s) << 32) |
                      (unsigned long long)(0xFFFFFFFFu - (unsigned)i);
    }
}

// ---------------- 4) one-block bitonic sort (desc) + gather top boxes ----------------
__global__ __launch_bounds__(1024)
void pg_sort_gather(const unsigned long long* __restrict__ cand,
                    const unsigned* __restrict__ meta,
                    const float4* __restrict__ proposals,
                    float4* __restrict__ topboxes) {
    __shared__ unsigned long long k[CAP];
    int tid = threadIdx.x;
    int count = (int)meta[0]; if (count > CAP) count = CAP;
    for (int i = tid; i < CAP; i += blockDim.x)
        k[i] = (i < count) ? cand[i] : 0ULL;
    __syncthreads();
    for (int kk = 2; kk <= CAP; kk <<= 1) {
        for (int j = kk >> 1; j > 0; j >>= 1) {
            for (int i = tid; i < CAP; i += blockDim.x) {
                int ixj = i ^ j;
                if (ixj > i) {
                    unsigned long long a = k[i], b = k[ixj];
                    if ((a < b) == ((i & kk) == 0)) { k[i] = b; k[ixj] = a; } // descending
                }
            }
            __syncthreads();
        }
    }
    int lim = count < PRE_NMS ? count : PRE_NMS;
    for (int i = tid; i < PAD; i += blockDim.x) {
        float4 bx = make_float4(0.f, 0.f, 0.f, 0.f);
        if (i < lim) {
            unsigned idx = 0xFFFFFFFFu - (unsigned)(k[i] & 0xFFFFFFFFull);
            bx = proposals[idx];
        }
        topboxes[i] = bx;
    }
}

// ---------------- 5) IoU suppression bitmask ----------------
// Block = 1 row-tile (16 rows) x 8 col-words (256 cols); one wave per col-word.
// Boxes staged in LDS via async global->LDS loads; WMMA computes area_i+area_j.
__global__ __launch_bounds__(256)
void pg_mask(const float4* __restrict__ topboxes,
             unsigned* __restrict__ mask) {
    __shared__ float4 ldsRow[16];
    __shared__ float4 ldsCol[256];

    int rt = blockIdx.x / CGRPS;          // row tile
    int cg = blockIdx.x % CGRPS;          // col-word group
    int r0 = rt * 16;
    int cbase = cg * 8 * 32;              // first column this block covers

    int t = threadIdx.x;
    int col = cbase + t; if (col > PAD - 1) col = PAD - 1;   // clamp (padded tail)
    async_copy_b128(&topboxes[col], &ldsCol[t]);
    if (t < 16)
        async_copy_b128(&topboxes[r0 + t], &ldsRow[t]);
    async_copy_wait();
    __syncthreads();

    int wv   = t >> 5;                    // wave id 0..7
    int cw   = cg * 8 + wv;               // global col-word
    if (cw >= WORDS) return;              // wave-uniform; no barriers below
    int lane = t & 31;
    bool lo  = lane < 16;
    int  n   = lane & 15;

    // per-lane row boxes for this lane's 8 output rows (M = r + (lo?0:8))
    float4 R[8];
#pragma unroll
    for (int r = 0; r < 8; ++r) R[r] = ldsRow[r + (lo ? 0 : 8)];
    // row areas for WMMA A operand: lanes 0..15 hold area(row r0+lane)
    float4 ra = ldsRow[n];
    float  rA = (ra.z - ra.x) * (ra.w - ra.y);

    unsigned wlo[8], whi[8];
#pragma unroll
    for (int s = 0; s < 2; ++s) {
        float4 cbx = ldsCol[wv * 32 + s * 16 + n];
        float  cA  = (cbx.z - cbx.x) * (cbx.w - cbx.y);

        // areaSum[M][N] = rA[M]*1 + 1*cA[N] via V_WMMA_F32_16X16X4_F32
        v2f A; A.x = lo ? rA   : 0.0f; A.y = lo ? 1.0f : 0.0f;
        v2f B; B.x = lo ? 1.0f : 0.0f; B.y = lo ? cA   : 0.0f;
        v8f C = {0.f,0.f,0.f,0.f,0.f,0.f,0.f,0.f};
        v8f S = __builtin_amdgcn_wmma_f32_16x16x4_f32(
                    false, A, false, B, (short)0, C, false, false);

#pragma unroll
        for (int r = 0; r < 8; ++r) {
            float iw = fmaxf(fminf(R[r].z, cbx.z) - fmaxf(R[r].x, cbx.x), 0.0f);
            float ih = fmaxf(fminf(R[r].w, cbx.w) - fmaxf(R[r].y, cbx.y), 0.0f);
            float inter = iw * ih;
            float uni   = S[r] - inter;              // area_i + area_j - inter
            // inter/uni > T  <=>  inter > T*uni  (uni>=0; uni==0 => inter==0)
            unsigned b = (unsigned)__ballot(inter > NMS_T * uni);
            if (s == 0) { wlo[r] = b & 0xFFFFu;          whi[r] = b >> 16; }
            else        { wlo[r] |= (b & 0xFFFFu) << 16; whi[r] |= (b >> 16) << 16; }
        }
    }
    if (lane == 0) {
#pragma unroll
        for (int r = 0; r < 8; ++r) {
            mask[(size_t)(r0 + r    ) * WORDS + cw] = wlo[r];
            mask[(size_t)(r0 + r + 8) * WORDS + cw] = whi[r];
        }
    }
}

// ---------------- 6) sequential greedy scan + scatter kept boxes ----------------
__global__ void pg_scan(const float* __restrict__ topboxes,
                        const unsigned* __restrict__ mask,
                        float* __restrict__ out) {
    __shared__ unsigned remv[WORDS];
    int tid = threadIdx.x;
    for (int w = tid; w < WORDS; w += blockDim.x) remv[w] = 0u;
    for (int i = tid; i < POST_NMS * 4; i += blockDim.x) out[i] = 0.0f;
    __syncthreads();
    int cnt = 0; // uniform across block (branches depend only on uniform shared data)
    for (int i = 0; i < PRE_NMS; ++i) {
        bool keep = ((remv[i >> 5] >> (i & 31)) & 1u) == 0u;
        if (keep) {
            if (cnt < POST_NMS && tid < 4)
                out[cnt * 4 + tid] = topboxes[i * 4 + tid];
            const unsigned* row = mask + (size_t)i * WORDS;
            for (int w = tid; w < WORDS; w += blockDim.x) remv[w] |= row[w];
            cnt++;
        }
        __syncthreads();
    }
}

// ---------------- launch ----------------
extern "C" void kernel_launch(void* const* d_in, const int* in_sizes, int n_in,
                              void* d_out, int out_size, void* d_ws, size_t ws_size,
                              hipStream_t stream) {
    const float* anchors = (const float*)d_in[1];
    const float* deltas  = (const float*)d_in[2];
    const float* logits  = (const float*)d_in[3];
    float* out = (float*)d_out;
    const int N = in_sizes[3]; // logits count == N anchors

    uint8_t* ws = (uint8_t*)d_ws;
    size_t oProp  = 0;
    size_t oScore = oProp  + (size_t)N * 16;
    size_t oHist  = oScore + (size_t)N * 4;
    size_t oMeta  = oHist  + (size_t)NBINS * 4;
    size_t oCand  = oMeta  + 64;
    size_t oTop   = oCand  + (size_t)CAP * 8;
    size_t oMask  = oTop   + (size_t)PAD * 16;

    float4*             proposals = (float4*)(ws + oProp);
    float*              scores    = (float*)(ws + oScore);
    unsigned*           hist      = (unsigned*)(ws + oHist);
    unsigned*           meta      = (unsigned*)(ws + oMeta);
    unsigned long long* cand      = (unsigned long long*)(ws + oCand);
    float4*             topboxes  = (float4*)(ws + oTop);
    unsigned*           maskbuf   = (unsigned*)(ws + oMask);

    pg_init<<<(NBINS + 255) / 256, 256, 0, stream>>>(hist, meta);
    pg_decode<<<(N + 255) / 256, 256, 0, stream>>>(anchors, deltas, logits,
                                                   proposals, scores, hist, N);
    pg_thresh<<<1, 32, 0, stream>>>(hist, meta);
    pg_compact<<<(N + 255) / 256, 256, 0, stream>>>(scores, meta, cand, N);
    pg_sort_gather<<<1, 1024, 0, stream>>>(cand, meta, proposals, topboxes);

    pg_mask<<<ROWTILES * CGRPS, 256, 0, stream>>>(topboxes, maskbuf);

    pg_scan<<<1, 256, 0, stream>>>((const float*)topboxes, maskbuf, out);
}